// HypergraphConv_74861279969932
// MI455X (gfx1250) — compile-verified
//
#include <hip/hip_runtime.h>

typedef float v2f __attribute__((ext_vector_type(2)));
typedef float v8f __attribute__((ext_vector_type(8)));

#define N_NODES 100000
#define N_EDGES 20000
#define FDIM    128

// ---------------------------------------------------------------------------
// Dense GEMM: Y[N x 128] = X[N x K] * W[K x 128], fp32 via V_WMMA_F32_16X16X4_F32.
// One wave computes one 16x16 output tile. Block = 256 threads = 8 waves,
// wave w handles column tile w (16 columns), so one block covers 16 rows x 128 cols.
//
// A-matrix 16x4 f32 VGPR layout (ISA 7.12.2):
//   lanes 0-15:  M=lane,    v0 = K=k+0, v1 = K=k+1
//   lanes 16-31: M=lane-16, v0 = K=k+2, v1 = K=k+3
// B-matrix 4x16 mirrored (N across lanes). C/D: VGPR r, lanes 0-15 -> (M=r, N=lane),
// lanes 16-31 -> (M=r+8, N=lane-16).
// ---------------------------------------------------------------------------
template <int K>
__global__ __launch_bounds__(256) void gemm_wmma_f32(const float* __restrict__ X,
                                                     const float* __restrict__ W,
                                                     float* __restrict__ Y)
{
    const int wave = threadIdx.x >> 5;   // 0..7 -> column tile
    const int lane = threadIdx.x & 31;
    const int l16  = lane & 15;
    const int hi   = lane >> 4;          // 0 or 1
    const int row0 = blockIdx.x * 16;
    const int col0 = wave * 16;

    const float* xrow = X + (size_t)(row0 + l16) * K;
    v8f acc = {};

#pragma unroll
    for (int k = 0; k < K; k += 4) {
        // A fragment: two contiguous floats -> single 8-byte load
        v2f a = *(const v2f*)(xrow + k + 2 * hi);
        // B fragment: W[k + 2*hi + {0,1}][col0 + l16]
        v2f b;
        b.x = W[(size_t)(k + 2 * hi + 0) * FDIM + col0 + l16];
        b.y = W[(size_t)(k + 2 * hi + 1) * FDIM + col0 + l16];
        acc = __builtin_amdgcn_wmma_f32_16x16x4_f32(
            /*neg_a=*/false, a, /*neg_b=*/false, b,
            /*c_mod=*/(short)0, acc, /*reuse_a=*/false, /*reuse_b=*/false);
    }

#pragma unroll
    for (int r = 0; r < 8; ++r) {
        const int m = r + hi * 8;
        Y[(size_t)(row0 + m) * FDIM + col0 + l16] = acc[r];
    }
}

// ---------------------------------------------------------------------------
// COO SpMM scatter: out[rows[e], :] += vals[e] * x[cols[e], :]
// One wave per nonzero; each lane owns 4 consecutive feature columns.
// Gathers are float4 (GLOBAL_LOAD_B128); scatter via global_atomic_add_f32.
// Both x and the accumulator are L2-resident (192 MB L2 >> 51.2 MB each).
// ---------------------------------------------------------------------------
__global__ __launch_bounds__(256) void spmm_scatter(const int* __restrict__ rows,
                                                    const int* __restrict__ cols,
                                                    const float* __restrict__ vals,
                                                    const float* __restrict__ X,
                                                    float* __restrict__ Y,
                                                    int nnz)
{
    const int gid  = blockIdx.x * blockDim.x + threadIdx.x;
    const int e    = gid >> 5;
    const int lane = threadIdx.x & 31;
    if (e >= nnz) return;

    const int   r = rows[e];
    const int   c = cols[e];
    const float v = vals[e];

    const float4 xv = ((const float4*)(X + (size_t)c * FDIM))[lane];
    float* yr = Y + (size_t)r * FDIM + lane * 4;
    atomicAdd(yr + 0, v * xv.x);
    atomicAdd(yr + 1, v * xv.y);
    atomicAdd(yr + 2, v * xv.z);
    atomicAdd(yr + 3, v * xv.w);
}

// ---------------------------------------------------------------------------
// Elementwise helpers (float4 wide).
// ---------------------------------------------------------------------------
__global__ void zero4_kernel(float4* __restrict__ p, size_t n4)
{
    size_t i = (size_t)blockIdx.x * blockDim.x + threadIdx.x;
    const size_t stride = (size_t)gridDim.x * blockDim.x;
    const float4 z = make_float4(0.f, 0.f, 0.f, 0.f);
    for (; i < n4; i += stride) p[i] = z;
}

__global__ void relu4_kernel(float4* __restrict__ p, size_t n4)
{
    size_t i = (size_t)blockIdx.x * blockDim.x + threadIdx.x;
    const size_t stride = (size_t)gridDim.x * blockDim.x;
    for (; i < n4; i += stride) {
        float4 v = p[i];
        v.x = fmaxf(v.x, 0.f);
        v.y = fmaxf(v.y, 0.f);
        v.z = fmaxf(v.z, 0.f);
        v.w = fmaxf(v.w, 0.f);
        p[i] = v;
    }
}

// ---------------------------------------------------------------------------
// Driver
// ---------------------------------------------------------------------------
extern "C" void kernel_launch(void* const* d_in, const int* in_sizes, int n_in,
                              void* d_out, int out_size, void* d_ws, size_t ws_size,
                              hipStream_t stream)
{
    (void)n_in; (void)out_size; (void)ws_size;

    const float* inputs = (const float*)d_in[0];   // 100000 x 256
    const float* W0     = (const float*)d_in[1];   // 256 x 128
    const float* W1     = (const float*)d_in[2];   // 128 x 128
    const float* W2     = (const float*)d_in[3];   // 128 x 128
    const int*   c1r    = (const int*)d_in[4];
    const int*   c1c    = (const int*)d_in[5];
    const float* c1v    = (const float*)d_in[6];
    const int*   c2r    = (const int*)d_in[7];
    const int*   c2c    = (const int*)d_in[8];
    const float* c2v    = (const float*)d_in[9];
    const int nnz1 = in_sizes[4];
    const int nnz2 = in_sizes[7];

    const size_t xElems = (size_t)N_NODES * FDIM;   // 12.8M floats
    const size_t yElems = (size_t)N_EDGES * FDIM;   // 2.56M floats

    float* out_x = (float*)d_out;            // final x (post-relu, layer 2)
    float* out_y = out_x + xElems;           // final y
    float* bufT  = (float*)d_ws;             // X @ W result (pre-spmm)
    float* bufX  = bufT + xElems;            // spmm + relu result

    const dim3 blk(256);
    const dim3 gemmGrid(N_NODES / 16);                       // 6250 row tiles
    const dim3 ewGrid(2048);
    auto spmmGrid = [](int nnz) { return dim3((unsigned)(((size_t)nnz * 32 + 255) / 256)); };

    // Layer 0: T = inputs @ W0 ; X = relu(spmm(coef1, T))
    gemm_wmma_f32<256><<<gemmGrid, blk, 0, stream>>>(inputs, W0, bufT);
    zero4_kernel<<<ewGrid, blk, 0, stream>>>((float4*)bufX, xElems / 4);
    spmm_scatter<<<spmmGrid(nnz1), blk, 0, stream>>>(c1r, c1c, c1v, bufT, bufX, nnz1);
    relu4_kernel<<<ewGrid, blk, 0, stream>>>((float4*)bufX, xElems / 4);

    // Layer 1: T = X @ W1 ; X = relu(spmm(coef1, T))
    gemm_wmma_f32<128><<<gemmGrid, blk, 0, stream>>>(bufX, W1, bufT);
    zero4_kernel<<<ewGrid, blk, 0, stream>>>((float4*)bufX, xElems / 4);
    spmm_scatter<<<spmmGrid(nnz1), blk, 0, stream>>>(c1r, c1c, c1v, bufT, bufX, nnz1);
    relu4_kernel<<<ewGrid, blk, 0, stream>>>((float4*)bufX, xElems / 4);

    // Layer 2: T = X @ W2 ; out_x = relu(spmm(coef1, T))
    gemm_wmma_f32<128><<<gemmGrid, blk, 0, stream>>>(bufX, W2, bufT);
    zero4_kernel<<<ewGrid, blk, 0, stream>>>((float4*)out_x, xElems / 4);
    spmm_scatter<<<spmmGrid(nnz1), blk, 0, stream>>>(c1r, c1c, c1v, bufT, out_x, nnz1);
    relu4_kernel<<<ewGrid, blk, 0, stream>>>((float4*)out_x, xElems / 4);

    // Final: out_y = relu(spmm(coef2, out_x))
    zero4_kernel<<<ewGrid, blk, 0, stream>>>((float4*)out_y, yElems / 4);
    spmm_scatter<<<spmmGrid(nnz2), blk, 0, stream>>>(c2r, c2c, c2v, out_x, out_y, nnz2);
    relu4_kernel<<<ewGrid, blk, 0, stream>>>((float4*)out_y, yElems / 4);
}